// GCN_26697516712417
// MI455X (gfx1250) — compile-verified
//
#include <hip/hip_runtime.h>
#include <hip/hip_bf16.h>
#include <math.h>

#define NN 8192
#define NE 131072
#define ENT (NN + NE)   // edges + self loops

typedef __attribute__((ext_vector_type(2))) float v2f;
typedef __attribute__((ext_vector_type(8))) float v8f;

// ---------------- Threefry-2x32 (JAX PRNG), host + device ----------------
__host__ __device__ inline unsigned rotl32(unsigned v, int d) {
  return (v << d) | (v >> (32 - d));
}

__host__ __device__ inline void threefry2x32(unsigned k0, unsigned k1,
                                             unsigned x0, unsigned x1,
                                             unsigned& o0, unsigned& o1) {
  const unsigned ks2 = k0 ^ k1 ^ 0x1BD11BDAu;
  const int R1[4] = {13, 15, 26, 6};
  const int R2[4] = {17, 29, 16, 24};
  x0 += k0; x1 += k1;
  for (int i = 0; i < 4; i++) { x0 += x1; x1 = rotl32(x1, R1[i]); x1 ^= x0; }
  x0 += k1; x1 += ks2 + 1u;
  for (int i = 0; i < 4; i++) { x0 += x1; x1 = rotl32(x1, R2[i]); x1 ^= x0; }
  x0 += ks2; x1 += k0 + 2u;
  for (int i = 0; i < 4; i++) { x0 += x1; x1 = rotl32(x1, R1[i]); x1 ^= x0; }
  x0 += k0; x1 += k1 + 3u;
  for (int i = 0; i < 4; i++) { x0 += x1; x1 = rotl32(x1, R2[i]); x1 ^= x0; }
  x0 += k1; x1 += ks2 + 4u;
  for (int i = 0; i < 4; i++) { x0 += x1; x1 = rotl32(x1, R1[i]); x1 ^= x0; }
  x0 += ks2; x1 += k0 + 5u;
  o0 = x0; o1 = x1;
}

// random_bits(key, 32, (n,)) element idx (n even), matching JAX's split-halves
__device__ inline unsigned rbits(unsigned k0, unsigned k1, unsigned idx, unsigned n) {
  unsigned half = n >> 1, a, b;
  if (idx < half) { threefry2x32(k0, k1, idx, idx + half, a, b); return a; }
  threefry2x32(k0, k1, idx - half, idx, a, b); return b;
}

__device__ inline float u01(unsigned bits) {   // [0,1) like JAX uniform
  return __uint_as_float((bits >> 9) | 0x3f800000u) - 1.0f;
}

// ---------------- float <-> order-preserving uint (for atomicMax) ----------------
__device__ inline unsigned ordf(float f) {
  unsigned u = __float_as_uint(f);
  return (u & 0x80000000u) ? ~u : (u | 0x80000000u);
}
__device__ inline float unordf(unsigned e) {
  unsigned u = (e & 0x80000000u) ? (e & 0x7FFFFFFFu) : ~e;
  return __uint_as_float(u);
}

// ---------------- GAT node linear: xl = x @ W.T; att scores ----------------
template <int IN, int H, int C>
__global__ void node_linear_kernel(const float* __restrict__ x, const float* __restrict__ W,
                                   const float* __restrict__ att_s, const float* __restrict__ att_d,
                                   float* __restrict__ xl, float* __restrict__ as_,
                                   float* __restrict__ ad_) {
  int n = blockIdx.x * blockDim.x + threadIdx.x;
  if (n >= NN) return;
  float xi[IN];
#pragma unroll
  for (int i = 0; i < IN; i++) xi[i] = x[n * IN + i];
  float v[H * C];
#pragma unroll
  for (int j = 0; j < H * C; j++) {
    float s = 0.f;
#pragma unroll
    for (int i = 0; i < IN; i++) s += xi[i] * W[j * IN + i];
    v[j] = s;
    xl[n * H * C + j] = s;
  }
#pragma unroll
  for (int h = 0; h < H; h++) {
    float ss = 0.f, sd = 0.f;
#pragma unroll
    for (int c = 0; c < C; c++) {
      ss += v[h * C + c] * att_s[h * C + c];
      sd += v[h * C + c] * att_d[h * C + c];
    }
    as_[n * H + h] = ss;
    ad_[n * H + h] = sd;
  }
}

// ---------------- GAT edge passes ----------------
template <int H>
__global__ void edge_score_kernel(const int* __restrict__ ei,
                                  const float* __restrict__ as_, const float* __restrict__ ad_,
                                  float* __restrict__ score, unsigned* __restrict__ emax) {
  int e = blockIdx.x * blockDim.x + threadIdx.x;
  if (e >= ENT) return;
  int s = (e < NE) ? ei[e] : (e - NE);
  int d = (e < NE) ? ei[NE + e] : (e - NE);
#pragma unroll
  for (int h = 0; h < H; h++) {
    float v = as_[s * H + h] + ad_[d * H + h];
    v = v > 0.f ? v : 0.2f * v;               // leaky_relu(0.2)
    score[e * H + h] = v;
    atomicMax(&emax[d * H + h], ordf(v));
  }
}

template <int H>
__global__ void edge_exp_kernel(const int* __restrict__ ei, float* __restrict__ score,
                                const unsigned* __restrict__ emax, float* __restrict__ den) {
  int e = blockIdx.x * blockDim.x + threadIdx.x;
  if (e >= ENT) return;
  int d = (e < NE) ? ei[NE + e] : (e - NE);
#pragma unroll
  for (int h = 0; h < H; h++) {
    float p = expf(score[e * H + h] - unordf(emax[d * H + h]));
    score[e * H + h] = p;
    atomicAdd(&den[d * H + h], p);
  }
}

template <int H, int C>
__global__ void edge_aggr_kernel(const int* __restrict__ ei, const float* __restrict__ score,
                                 const float* __restrict__ den, const float* __restrict__ xl,
                                 float* __restrict__ agg) {
  int e = blockIdx.x * blockDim.x + threadIdx.x;
  if (e >= ENT) return;
  int s = (e < NE) ? ei[e] : (e - NE);
  int d = (e < NE) ? ei[NE + e] : (e - NE);
#pragma unroll
  for (int h = 0; h < H; h++) {
    float alpha = score[e * H + h] / den[d * H + h];
#pragma unroll
    for (int c = 0; c < C; c++)
      atomicAdd(&agg[(d * H + h) * C + c], alpha * xl[(s * H + h) * C + c]);
  }
}

// ---------------- bias + relu + dropout (exact JAX bernoulli) ----------------
__global__ void post_kernel(const float* __restrict__ agg, const float* __restrict__ bias,
                            float* __restrict__ xout, int HC, int doRelu, int doDrop,
                            unsigned k0, unsigned k1) {
  int i = blockIdx.x * blockDim.x + threadIdx.x;
  int n = NN * HC;
  if (i >= n) return;
  float v = agg[i] + bias[i % HC];
  if (doRelu) v = fmaxf(v, 0.f);
  if (doDrop) {
    unsigned half = (unsigned)(n >> 1), a, b, bits;
    if ((unsigned)i < half) { threefry2x32(k0, k1, (unsigned)i, (unsigned)i + half, a, b); bits = a; }
    else { threefry2x32(k0, k1, (unsigned)i - half, (unsigned)i, a, b); bits = b; }
    v = (u01(bits) < 0.5f) ? v * 2.0f : 0.0f;
  }
  xout[i] = v;
}

// ---------------- fp32 WMMA GEMV: y[row] = sum_k W[row,k]*x[k] + bias[row] ----------------
// One wave per 16-row block. B tile = W[rows, k0..k0+3] (4x16 K-striped across lanes),
// A = x chunk broadcast over M; every D row is identical, lanes 0..15 of acc[0] = outputs.
__global__ void __launch_bounds__(256)
gemv_wmma_kernel(const float* __restrict__ W, const float* __restrict__ x,
                 const float* __restrict__ bias, float* __restrict__ y, int K) {
  const int wave = threadIdx.x >> 5;
  const int lane = threadIdx.x & 31;
  const int rowBase = (blockIdx.x * 8 + wave) << 4;
  const int r = lane & 15;
  const int koff = (lane >> 4) << 1;           // lanes 0-15: K 0,1 ; lanes 16-31: K 2,3
  const float* wrow = W + (size_t)(rowBase + r) * (size_t)K + koff;
  const float* xp = x + koff;
  v8f acc = {0.f, 0.f, 0.f, 0.f, 0.f, 0.f, 0.f, 0.f};
#pragma unroll 4
  for (int k0 = 0; k0 < K; k0 += 4) {
    __builtin_prefetch(wrow + k0 + 2048, 0, 3);            // global_prefetch_b8, 8KB ahead
    v2f a = *reinterpret_cast<const v2f*>(xp + k0);        // A[m,k]=x[k0+k] broadcast
    v2f b = *reinterpret_cast<const v2f*>(wrow + k0);      // B[k,n]=W[rowBase+n, k0+k]
    acc = __builtin_amdgcn_wmma_f32_16x16x4_f32(false, a, false, b, (short)0, acc, false, false);
  }
  if (lane < 16) {
    int row = rowBase + lane;
    y[row] = acc[0] + bias[row];
  }
}

// ---------------- value head: out[8192] = dot(y1, c2_w) + c2_b ----------------
__global__ void __launch_bounds__(1024)
value_kernel(const float* __restrict__ y1, const float* __restrict__ c2w,
             const float* __restrict__ c2b, float* __restrict__ out) {
  __shared__ float red[1024];
  int t = threadIdx.x;
  float s = 0.f;
  for (int i = t; i < NN; i += 1024) s += y1[i] * c2w[i];
  red[t] = s; __syncthreads();
  for (int off = 512; off > 0; off >>= 1) {
    if (t < off) red[t] += red[t + off];
    __syncthreads();
  }
  if (t == 0) out[NN] = red[0] + c2b[0];
}

// ---------------- policy head: mask+tanh, softmax -> out[0..8191], gumbel argmax -> out[8193]
__global__ void __launch_bounds__(1024)
policy_kernel(const float* __restrict__ h, const unsigned char* __restrict__ mask,
              float* __restrict__ out, unsigned ka0, unsigned ka1) {
  __shared__ float red[1024];
  __shared__ int ired[1024];
  const int t = threadIdx.x;
  float pv[8];
#pragma unroll
  for (int j = 0; j < 8; j++) {
    int i = t + j * 1024;
    pv[j] = mask[i] ? tanhf(h[i]) : -999999.0f;
  }
  float m = pv[0];
#pragma unroll
  for (int j = 1; j < 8; j++) m = fmaxf(m, pv[j]);
  red[t] = m; __syncthreads();
  for (int off = 512; off > 0; off >>= 1) {
    if (t < off) red[t] = fmaxf(red[t], red[t + off]);
    __syncthreads();
  }
  float gmax = red[0]; __syncthreads();
  float s = 0.f;
#pragma unroll
  for (int j = 0; j < 8; j++) s += expf(pv[j] - gmax);
  red[t] = s; __syncthreads();
  for (int off = 512; off > 0; off >>= 1) {
    if (t < off) red[t] += red[t + off];
    __syncthreads();
  }
  float inv = 1.0f / red[0]; __syncthreads();
  float best = -3.4e38f; int bi = 0;
#pragma unroll
  for (int j = 0; j < 8; j++) {
    int i = t + j * 1024;
    out[i] = expf(pv[j] - gmax) * inv;
    float u = u01(rbits(ka0, ka1, (unsigned)i, (unsigned)NN));
    u = u * (1.0f - 1.1754944e-38f) + 1.1754944e-38f;    // uniform(tiny, 1)
    float z = pv[j] - logf(-logf(u));                    // gumbel perturbation
    if (z > best || (z == best && i < bi)) { best = z; bi = i; }
  }
  red[t] = best; ired[t] = bi; __syncthreads();
  for (int off = 512; off > 0; off >>= 1) {
    if (t < off) {
      float ov = red[t + off]; int oi = ired[t + off];
      if (ov > red[t] || (ov == red[t] && oi < ired[t])) { red[t] = ov; ired[t] = oi; }
    }
    __syncthreads();
  }
  if (t == 0) out[NN + 1] = (float)ired[0];
}

// ---------------- host ----------------
extern "C" void kernel_launch(void* const* d_in, const int* in_sizes, int n_in,
                              void* d_out, int out_size, void* d_ws, size_t ws_size,
                              hipStream_t stream) {
  const float* data = (const float*)d_in[0];
  const int* ei = (const int*)d_in[1];
  const unsigned char* mask = (const unsigned char*)d_in[3];
  const float* W1 = (const float*)d_in[4];  const float* b1 = (const float*)d_in[5];
  const float* as1 = (const float*)d_in[6]; const float* ad1 = (const float*)d_in[7];
  const float* W2 = (const float*)d_in[8];  const float* b2 = (const float*)d_in[9];
  const float* as2 = (const float*)d_in[10]; const float* ad2 = (const float*)d_in[11];
  const float* W3 = (const float*)d_in[12]; const float* b3 = (const float*)d_in[13];
  const float* as3 = (const float*)d_in[14]; const float* ad3 = (const float*)d_in[15];
  const float* l1a_w = (const float*)d_in[16]; const float* l1a_b = (const float*)d_in[17];
  const float* l1b_w = (const float*)d_in[18]; const float* l1b_b = (const float*)d_in[19];
  const float* c1_w = (const float*)d_in[20]; const float* c1_b = (const float*)d_in[21];
  const float* c2_w = (const float*)d_in[22]; const float* c2_b = (const float*)d_in[23];
  float* out = (float*)d_out;
  float* ws = (float*)d_ws;

  // JAX key(42) -> split into (k1, k2, ka): threefry over counts [0..5]
  unsigned A0, B0, A1, B1, A2, B2;
  threefry2x32(0u, 42u, 0u, 3u, A0, B0);
  threefry2x32(0u, 42u, 1u, 4u, A1, B1);
  threefry2x32(0u, 42u, 2u, 5u, A2, B2);
  const unsigned k1a = A0, k1b = A1, k2a = A2, k2b = B0, kaa = B1, kab = B2;

  // workspace layout (floats)
  float* xl = ws + 0;                  // 32768
  float* as_ = ws + 32768;             // 16384
  float* ad_ = ws + 49152;             // 16384
  unsigned* emax = (unsigned*)(ws + 65536);  // 16384
  float* den = ws + 81920;             // 16384
  float* score = ws + 98304;           // 278528 (ENT * 2)
  float* agg = ws + 376832;            // 32768
  float* xn = ws + 409600;             // 32768 (node feats / final x)
  float* h1 = ws + 442368;             // 16384
  float* hv = ws + 458752;             // 8192
  float* y1 = ws + 466944;             // 8192

  const int eBlk = (ENT + 255) / 256;

  // ---- GAT layer 1: 3 -> (2 heads x 2) ----
  node_linear_kernel<3, 2, 2><<<32, 256, 0, stream>>>(data, W1, as1, ad1, xl, as_, ad_);
  hipMemsetAsync(emax, 0, NN * 2 * sizeof(unsigned), stream);
  hipMemsetAsync(den, 0, NN * 2 * sizeof(float), stream);
  hipMemsetAsync(agg, 0, NN * 4 * sizeof(float), stream);
  edge_score_kernel<2><<<eBlk, 256, 0, stream>>>(ei, as_, ad_, score, emax);
  edge_exp_kernel<2><<<eBlk, 256, 0, stream>>>(ei, score, emax, den);
  edge_aggr_kernel<2, 2><<<eBlk, 256, 0, stream>>>(ei, score, den, xl, agg);
  post_kernel<<<(NN * 4 + 255) / 256, 256, 0, stream>>>(agg, b1, xn, 4, 1, 1, k1a, k1b);

  // ---- GAT layer 2: 4 -> (2 heads x 1) ----
  node_linear_kernel<4, 2, 1><<<32, 256, 0, stream>>>(xn, W2, as2, ad2, xl, as_, ad_);
  hipMemsetAsync(emax, 0, NN * 2 * sizeof(unsigned), stream);
  hipMemsetAsync(den, 0, NN * 2 * sizeof(float), stream);
  hipMemsetAsync(agg, 0, NN * 2 * sizeof(float), stream);
  edge_score_kernel<2><<<eBlk, 256, 0, stream>>>(ei, as_, ad_, score, emax);
  edge_exp_kernel<2><<<eBlk, 256, 0, stream>>>(ei, score, emax, den);
  edge_aggr_kernel<2, 1><<<eBlk, 256, 0, stream>>>(ei, score, den, xl, agg);
  post_kernel<<<(NN * 2 + 255) / 256, 256, 0, stream>>>(agg, b2, xn, 2, 1, 1, k2a, k2b);

  // ---- GAT layer 3: 2 -> (1 head x 1), no relu/dropout ----
  node_linear_kernel<2, 1, 1><<<32, 256, 0, stream>>>(xn, W3, as3, ad3, xl, as_, ad_);
  hipMemsetAsync(emax, 0, NN * sizeof(unsigned), stream);
  hipMemsetAsync(den, 0, NN * sizeof(float), stream);
  hipMemsetAsync(agg, 0, NN * sizeof(float), stream);
  edge_score_kernel<1><<<eBlk, 256, 0, stream>>>(ei, as_, ad_, score, emax);
  edge_exp_kernel<1><<<eBlk, 256, 0, stream>>>(ei, score, emax, den);
  edge_aggr_kernel<1, 1><<<eBlk, 256, 0, stream>>>(ei, score, den, xl, agg);
  post_kernel<<<32, 256, 0, stream>>>(agg, b3, xn, 1, 0, 0, 0u, 0u);

  // ---- dense heads (memory-bound GEMVs, fp32 WMMA) ----
  gemv_wmma_kernel<<<64, 256, 0, stream>>>(c1_w, xn, c1_b, y1, 8192);     // 8192x8192
  gemv_wmma_kernel<<<128, 256, 0, stream>>>(l1a_w, xn, l1a_b, h1, 8192);  // 16384x8192
  gemv_wmma_kernel<<<64, 256, 0, stream>>>(l1b_w, h1, l1b_b, hv, 16384);  // 8192x16384

  value_kernel<<<1, 1024, 0, stream>>>(y1, c2_w, c2_b, out);
  policy_kernel<<<1, 1024, 0, stream>>>(hv, mask, out, kaa, kab);
}